// QSA_35837207118674
// MI455X (gfx1250) — compile-verified
//
#include <hip/hip_runtime.h>
#include <math.h>

// ---------------------------------------------------------------------------
// Quantum circuit expectation via precomputed unitary + f32 WMMA GEMM.
//   out[n,w] = sum_i |(U x_norm[n])_i|^2 * (-1)^{bit_w(i)}
// Kernel 1: build B[k][i] = U[i][k] (real & imag, f32) by evolving basis states.
// Kernel 2: psi = Xn * B (two real GEMMs via V_WMMA_F32_16X16X4_F32),
//           fused normalization + |psi|^2 + Z-sign reduction.
// Workspace: 2 * 1024*1024 * 4 B = 8 MB.
// ---------------------------------------------------------------------------

#define NW 10
#define DIM 1024

typedef float v2f __attribute__((ext_vector_type(2)));
typedef float v8f __attribute__((ext_vector_type(8)));

// -------------------------- Kernel 1: build unitary -------------------------
// Block k evolves basis state |k>; result column U[:,k] = row k of B.
__global__ __launch_bounds__(256) void QSA_build_unitary(
    const float* __restrict__ rx0, const float* __restrict__ ry0,
    const float* __restrict__ ry1, float* __restrict__ Br,
    float* __restrict__ Bi) {
  __shared__ float vr[DIM];
  __shared__ float vi[DIM];
  const int k = blockIdx.x;
  const int tid = threadIdx.x;

  for (int i = tid; i < DIM; i += 256) {
    vr[i] = (i == k) ? 1.0f : 0.0f;
    vi[i] = 0.0f;
  }
  __syncthreads();

  // Layer 1: RX(rx0[j]) then RY(ry0[j]) on each wire (wire 0 = MSB).
  for (int j = 0; j < NW; ++j) {
    const unsigned mask = 1u << (9 - j);
    const unsigned lowm = mask - 1u;
    // RX: [c, -i s; -i s, c]
    {
      const float h = 0.5f * rx0[j];
      const float c = cosf(h), s = sinf(h);
      for (int p = tid; p < DIM / 2; p += 256) {
        const unsigned i0 = ((p & ~lowm) << 1) | (p & lowm);
        const unsigned i1 = i0 | mask;
        const float r0 = vr[i0], m0 = vi[i0], r1 = vr[i1], m1 = vi[i1];
        vr[i0] = c * r0 + s * m1;
        vi[i0] = c * m0 - s * r1;
        vr[i1] = c * r1 + s * m0;
        vi[i1] = c * m1 - s * r0;
      }
      __syncthreads();
    }
    // RY: [c, -s; s, c]
    {
      const float h = 0.5f * ry0[j];
      const float c = cosf(h), s = sinf(h);
      for (int p = tid; p < DIM / 2; p += 256) {
        const unsigned i0 = ((p & ~lowm) << 1) | (p & lowm);
        const unsigned i1 = i0 | mask;
        const float r0 = vr[i0], m0 = vi[i0], r1 = vr[i1], m1 = vi[i1];
        vr[i0] = c * r0 - s * r1;
        vi[i0] = c * m0 - s * m1;
        vr[i1] = s * r0 + c * r1;
        vi[i1] = s * m0 + c * m1;
      }
      __syncthreads();
    }
  }

  // CNOT ring: control j, target (j+1)%10, applied sequentially.
  for (int j = 0; j < NW; ++j) {
    const unsigned cmask = 1u << (9 - j);
    const unsigned tmask = 1u << (9 - ((j + 1) % NW));
    for (int i = tid; i < DIM; i += 256) {
      if ((i & cmask) && !(i & tmask)) {
        const unsigned ip = i | tmask;
        const float tr = vr[i], ti = vi[i];
        vr[i] = vr[ip]; vi[i] = vi[ip];
        vr[ip] = tr;    vi[ip] = ti;
      }
    }
    __syncthreads();
  }

  // Layer 2: RY(ry1[j]).
  for (int j = 0; j < NW; ++j) {
    const unsigned mask = 1u << (9 - j);
    const unsigned lowm = mask - 1u;
    const float h = 0.5f * ry1[j];
    const float c = cosf(h), s = sinf(h);
    for (int p = tid; p < DIM / 2; p += 256) {
      const unsigned i0 = ((p & ~lowm) << 1) | (p & lowm);
      const unsigned i1 = i0 | mask;
      const float r0 = vr[i0], m0 = vi[i0], r1 = vr[i1], m1 = vi[i1];
      vr[i0] = c * r0 - s * r1;
      vi[i0] = c * m0 - s * m1;
      vr[i1] = s * r0 + c * r1;
      vi[i1] = s * m0 + c * m1;
    }
    __syncthreads();
  }

  // Row k of B (coalesced).
  for (int i = tid; i < DIM; i += 256) {
    Br[(size_t)k * DIM + i] = vr[i];
    Bi[(size_t)k * DIM + i] = vi[i];
  }
}

// ---------------- Kernel 2: normalize + WMMA GEMM + Z reduce -----------------
// Block = 16 rows of X, 8 waves; wave wv owns output columns [wv*128, wv*128+128).
// LDS A-strip XOR-swizzled: element (r,c) at (r<<10) | (c ^ (r<<2)).
__device__ __forceinline__ int swz(int r, int c) {
  return (r << 10) | (c ^ (r << 2));
}

__global__ __launch_bounds__(256) void QSA_gemm_z(
    const float* __restrict__ X, const float* __restrict__ Br,
    const float* __restrict__ Bi, float* __restrict__ out) {
  __shared__ float As[16 * DIM];  // 64 KB
  const int tid = threadIdx.x;
  const int n0 = blockIdx.x * 16;

  // Zero this block's exclusive output slice (harness poisons d_out).
  if (tid < 16 * NW) out[(size_t)n0 * NW + tid] = 0.0f;

  // Stage 16 x 1024 strip of X (coalesced global reads).
  for (int idx = tid; idx < 16 * DIM; idx += 256) {
    const int r = idx >> 10, c = idx & (DIM - 1);
    As[swz(r, c)] = X[((size_t)(n0 + r) << 10) | c];
  }
  __syncthreads();

  // Normalize rows in LDS: wave wv handles rows wv and wv+8.
  const int wv = tid >> 5;
  const int lane = tid & 31;
  for (int r = wv; r < 16; r += 8) {
    float ss = 0.0f;
    for (int c = lane; c < DIM; c += 32) {
      const float v = As[swz(r, c)];
      ss += v * v;
    }
    for (int off = 16; off; off >>= 1) ss += __shfl_xor(ss, off, 32);
    const float inv = 1.0f / fmaxf(sqrtf(ss), 1e-12f);
    for (int c = lane; c < DIM; c += 32) As[swz(r, c)] *= inv;
  }
  __syncthreads();

  // WMMA fragment coordinates (V_WMMA_F32_16X16X4_F32, wave32):
  //   A 16x4 : lane m = lane&15, VGPR v = K (half*2 + v)
  //   B 4x16 : lane col = lane&15, VGPR v = K (half*2 + v)
  //   D 16x16: VGPR v -> row v + 8*half, col = lane&15
  const int m = lane & 15;
  const int hgrp = lane >> 4;
  const int kofs = hgrp * 2;

  float zacc[8][NW];
#pragma unroll
  for (int v = 0; v < 8; ++v)
#pragma unroll
    for (int w = 0; w < NW; ++w) zacc[v][w] = 0.0f;

  for (int tt = 0; tt < 8; ++tt) {
    const int col = (wv * 8 + tt) * 16 + m;
    const float* pBr = Br + col;
    const float* pBi = Bi + col;
    v8f accR = {};
    v8f accI = {};
    for (int k = 0; k < DIM; k += 4) {
      const int k2 = k + kofs;
      // A pair is contiguous in LDS (swizzle preserves even pairs) -> b64 load.
      const v2f a = *(const v2f*)&As[swz(m, k2)];
      v2f br, bi;
      br.x = pBr[(size_t)k2 << 10];
      br.y = pBr[(size_t)(k2 + 1) << 10];
      bi.x = pBi[(size_t)k2 << 10];
      bi.y = pBi[(size_t)(k2 + 1) << 10];
      accR = __builtin_amdgcn_wmma_f32_16x16x4_f32(false, a, false, br,
                                                   (short)0, accR, false, false);
      accI = __builtin_amdgcn_wmma_f32_16x16x4_f32(false, a, false, bi,
                                                   (short)0, accI, false, false);
    }
    // Fused |psi|^2 * Z-sign accumulate; column (hence signs) fixed per lane.
    float sgn[NW];
#pragma unroll
    for (int w = 0; w < NW; ++w)
      sgn[w] = ((col >> (9 - w)) & 1) ? -1.0f : 1.0f;
#pragma unroll
    for (int v = 0; v < 8; ++v) {
      const float p = accR[v] * accR[v] + accI[v] * accI[v];
#pragma unroll
      for (int w = 0; w < NW; ++w) zacc[v][w] += p * sgn[w];
    }
  }

  // Reduce across the 16 columns held by a half-wave, then accumulate.
#pragma unroll
  for (int v = 0; v < 8; ++v) {
#pragma unroll
    for (int w = 0; w < NW; ++w) {
      float t = zacc[v][w];
      t += __shfl_xor(t, 1, 16);
      t += __shfl_xor(t, 2, 16);
      t += __shfl_xor(t, 4, 16);
      t += __shfl_xor(t, 8, 16);
      if (m == 0)
        atomicAdd(&out[(size_t)(n0 + v + 8 * hgrp) * NW + w], t);
    }
  }
}

// ------------------------------- launcher -----------------------------------
extern "C" void kernel_launch(void* const* d_in, const int* in_sizes, int n_in,
                              void* d_out, int out_size, void* d_ws,
                              size_t ws_size, hipStream_t stream) {
  (void)in_sizes; (void)n_in; (void)out_size; (void)ws_size;
  const float* x   = (const float*)d_in[0];  // (4,1024,1024) -> (4096,1024)
  const float* rx0 = (const float*)d_in[1];
  const float* ry0 = (const float*)d_in[2];
  const float* ry1 = (const float*)d_in[3];
  float* out = (float*)d_out;

  float* Br = (float*)d_ws;               // 4 MB
  float* Bi = Br + (size_t)DIM * DIM;     // 4 MB

  QSA_build_unitary<<<DIM, 256, 0, stream>>>(rx0, ry0, ry1, Br, Bi);
  QSA_gemm_z<<<4096 / 16, 256, 0, stream>>>(x, Br, Bi, out);
}